// HQLayer_38749194945207
// MI455X (gfx1250) — compile-verified
//
#include <hip/hip_runtime.h>

typedef float v2f __attribute__((ext_vector_type(2)));
typedef float v8f __attribute__((ext_vector_type(8)));

#define NQ 4
#define IN_DIM 256
#define OUT_DIM 64
#define W1_STRIDE 260   // 16-row padded LDS stride; 260%64=4 -> conflict-free banks

// Apply a generic complex 2x2 gate on the qubit selected by `bit`.
// Called only with compile-time-constant `bit` after full unrolling.
__device__ __forceinline__ void apply1q(float ar[16], float ai[16], int bit,
                                        float u00r, float u00i, float u01r, float u01i,
                                        float u10r, float u10i, float u11r, float u11i) {
#pragma unroll
  for (int i = 0; i < 16; ++i) {
    if ((i & bit) == 0) {
      const int j = i | bit;
      float r0 = ar[i], q0 = ai[i], r1 = ar[j], q1 = ai[j];
      ar[i] = u00r * r0 - u00i * q0 + u01r * r1 - u01i * q1;
      ai[i] = u00r * q0 + u00i * r0 + u01r * q1 + u01i * r1;
      ar[j] = u10r * r0 - u10i * q0 + u11r * r1 - u11i * q1;
      ai[j] = u10r * q0 + u10i * r0 + u11r * q1 + u11i * r1;
    }
  }
}

__device__ __forceinline__ void cnot_gate(float ar[16], float ai[16], int cbit, int tbit) {
#pragma unroll
  for (int i = 0; i < 16; ++i) {
    if ((i & cbit) != 0 && (i & tbit) == 0) {
      const int j = i | tbit;
      float tr = ar[i]; ar[i] = ar[j]; ar[j] = tr;
      float ti = ai[i]; ai[i] = ai[j]; ai[j] = ti;
    }
  }
}

__global__ __launch_bounds__(256) void hq_fused(
    const float* __restrict__ x, const float* __restrict__ W1,
    const float* __restrict__ b1, const float* __restrict__ qw,
    const float* __restrict__ W2, const float* __restrict__ b2,
    float* __restrict__ out) {
  __shared__ float lds_w1[16 * W1_STRIDE];  // W1^T padded: rows>=4 are zero
  __shared__ float lds_a[8 * 64];           // per-wave a-matrix exchange
  __shared__ float lds_rot[8 * 8];          // 8 Rot gates x 8 matrix entries (shared by batch)

  const int tid = threadIdx.x;
  // Stage W1 into LDS, zero-padded to 16 "N" rows so WMMA B-lanes 4..15 read 0.
  for (int i = tid; i < 16 * W1_STRIDE; i += 256) {
    int r = i / W1_STRIDE, c = i - r * W1_STRIDE;
    lds_w1[i] = (r < NQ && c < IN_DIM) ? W1[r * IN_DIM + c] : 0.0f;
  }
  // Precompute the 8 sample-independent Rot matrices ONCE per block (full-precision trig).
  // Rot = [[e^{-i(p2+o2)}ct, -e^{i(p2-o2)}st], [e^{-i(p2-o2)}st, e^{i(p2+o2)}ct]]
  if (tid < 2 * NQ) {
    const float* g = qw + tid * 3;
    float p2 = 0.5f * g[0], t2 = 0.5f * g[1], o2 = 0.5f * g[2];
    float ct = cosf(t2), st = sinf(t2);
    float ca = cosf(p2 + o2), sa = sinf(p2 + o2);
    float cb = cosf(p2 - o2), sb = sinf(p2 - o2);
    float* r = lds_rot + tid * 8;
    r[0] = ca * ct;  r[1] = -sa * ct;   // u00
    r[2] = -cb * st; r[3] = -sb * st;   // u01
    r[4] = cb * st;  r[5] = -sb * st;   // u10
    r[6] = ca * ct;  r[7] = sa * ct;    // u11
  }
  __syncthreads();

  const int wave = tid >> 5;
  const int lane = tid & 31;
  const int hi   = lane >> 4;    // half-wave select (K-offset for A/B fragments)
  const int lo   = lane & 15;
  const int koff = hi * 2;

  const int tile = blockIdx.x * 8 + wave;   // 16 samples per wave
  const int row0 = tile * 16;

  // ---- GEMM1: a[16x4] = x_tile[16x256] @ W1^T[256x4], via 64 wmma f32 16x16x4 ----
  v8f acc = {};
  const float* xp = x + (size_t)(row0 + lo) * IN_DIM + koff;
  const float* wp = lds_w1 + lo * W1_STRIDE + koff;
#pragma unroll 8
  for (int k = 0; k < IN_DIM; k += 4) {
    v2f av = *(const v2f*)(xp + k);   // A[M=lo][k+koff .. +1]
    v2f bv = *(const v2f*)(wp + k);   // B[k+koff..][N=lo] = W1[lo][k+koff..]
    acc = __builtin_amdgcn_wmma_f32_16x16x4_f32(false, av, false, bv,
                                                (short)0, acc, false, false);
  }

  // ---- Transpose a-matrix to lanes (only columns 0..3 are meaningful) ----
  float* my = lds_a + wave * 64;
  if (lo < NQ) {
#pragma unroll
    for (int r = 0; r < 8; ++r) my[(hi * 8 + r) * NQ + lo] = acc[r];
  }
  __syncthreads();

  float a0 = my[lo * NQ + 0] + b1[0];
  float a1 = my[lo * NQ + 1] + b1[1];
  float a2 = my[lo * NQ + 2] + b1[2];
  float a3 = my[lo * NQ + 3] + b1[3];

  // ---- 4-qubit statevector sim in registers (lanes m and 16+m redundantly do sample m;
  //      redundancy is free at wave level and avoids a second LDS shuffle) ----
  float ar[16], ai[16];
#pragma unroll
  for (int i = 0; i < 16; ++i) { ar[i] = 0.0f; ai[i] = 0.0f; }
  ar[0] = 1.0f;

  float ang[NQ] = {a0, a1, a2, a3};
#pragma unroll
  for (int w = 0; w < NQ; ++w) {   // AngleEmbedding: RX(ang[w]) on wire w (fast native trig)
    float ch = __cosf(0.5f * ang[w]);
    float sh = __sinf(0.5f * ang[w]);
    apply1q(ar, ai, 8 >> w, ch, 0.0f, 0.0f, -sh, 0.0f, -sh, ch, 0.0f);
  }
#pragma unroll
  for (int l = 0; l < 2; ++l) {    // StronglyEntanglingLayers
#pragma unroll
    for (int w = 0; w < NQ; ++w) {
      const float* g = lds_rot + (l * NQ + w) * 8;  // broadcast LDS reads
      apply1q(ar, ai, 8 >> w,
              g[0], g[1], g[2], g[3], g[4], g[5], g[6], g[7]);
    }
    const int rr = (l % 3) + 1;    // ring range: l=0 -> 1, l=1 -> 2
#pragma unroll
    for (int w = 0; w < NQ; ++w) {
      const int t = (w + rr) & 3;
      cnot_gate(ar, ai, 8 >> w, 8 >> t);
    }
  }
  // <Z_w> expectations (probabilities computed once, 4 signed reductions)
  float pr[16];
#pragma unroll
  for (int i = 0; i < 16; ++i) pr[i] = ar[i] * ar[i] + ai[i] * ai[i];
  float q[NQ];
#pragma unroll
  for (int w = 0; w < NQ; ++w) {
    const int bit = 8 >> w;
    float e = 0.0f;
#pragma unroll
    for (int i = 0; i < 16; ++i) e += (i & bit) ? -pr[i] : pr[i];
    q[w] = e;
  }

  // ---- GEMM2: out[16x64] = q[16x4] @ W2^T[4x64] + b2, four wmma N-tiles of 16 ----
  v2f af;
  af.x = hi ? q[2] : q[0];   // A[M=lo][K=koff]
  af.y = hi ? q[3] : q[1];   // A[M=lo][K=koff+1]
#pragma unroll
  for (int t = 0; t < 4; ++t) {
    const int col = t * 16 + lo;
    v2f bv = *(const v2f*)(W2 + col * NQ + koff);  // B[k][col] = W2[col][k]
    const float bias = b2[col];
    v8f cv = {bias, bias, bias, bias, bias, bias, bias, bias};
    v8f d = __builtin_amdgcn_wmma_f32_16x16x4_f32(false, af, false, bv,
                                                  (short)0, cv, false, false);
    float* orow = out + (size_t)(row0 + hi * 8) * OUT_DIM + col;
#pragma unroll
    for (int r = 0; r < 8; ++r) orow[r * OUT_DIM] = d[r];
  }
}

extern "C" void kernel_launch(void* const* d_in, const int* in_sizes, int n_in,
                              void* d_out, int out_size, void* d_ws, size_t ws_size,
                              hipStream_t stream) {
  (void)n_in; (void)out_size; (void)d_ws; (void)ws_size;
  const float* x  = (const float*)d_in[0];
  const float* W1 = (const float*)d_in[1];
  const float* b1 = (const float*)d_in[2];
  const float* qw = (const float*)d_in[3];
  const float* W2 = (const float*)d_in[4];
  const float* b2 = (const float*)d_in[5];
  float* out = (float*)d_out;

  const int batch  = in_sizes[0] / IN_DIM;   // 262144
  const int blocks = batch / 128;            // 16 samples/wave * 8 waves/block
  hq_fused<<<blocks, 256, 0, stream>>>(x, W1, b1, qw, W2, b2, out);
}